// GCNModel_11897059410631
// MI455X (gfx1250) — compile-verified
//
#include <hip/hip_runtime.h>

#define NN   100000
#define NE   3200000
#define NG   1024
#define HID  64

typedef __attribute__((ext_vector_type(2))) float v2f;
typedef __attribute__((ext_vector_type(8))) float v8f;

// ---------- degree / norm precompute ----------
__global__ void k_init_deg(float* deg, int n) {
    int i = blockIdx.x * blockDim.x + threadIdx.x;
    if (i < n) deg[i] = 1.0f;                       // self-loop contributes 1
}

__global__ void k_deg_accum(const int* __restrict__ dst, float* deg, int e) {
    int i = blockIdx.x * blockDim.x + threadIdx.x;
    if (i < e) atomicAdd(&deg[dst[i]], 1.0f);
}

__global__ void k_dinv(float* deg, int n) {
    int i = blockIdx.x * blockDim.x + threadIdx.x;
    if (i < n) {
        float d = deg[i];
        deg[i] = (d > 0.0f) ? rsqrtf(d) : 0.0f;
    }
}

__global__ void k_norm(const int* __restrict__ src, const int* __restrict__ dst,
                       const float* __restrict__ dinv, float* __restrict__ norm, int e) {
    int i = blockIdx.x * blockDim.x + threadIdx.x;
    if (i < e) norm[i] = dinv[src[i]] * dinv[dst[i]];
}

// ---------- layer 1 transform: t = x (N,1) outer w1 (1,H) ----------
__global__ void k_outer1(const float* __restrict__ x, const float* __restrict__ w1,
                         float* __restrict__ t, int total) {
    int tid = blockIdx.x * blockDim.x + threadIdx.x;
    if (tid < total) {
        int i = tid >> 6, j = tid & 63;
        t[tid] = x[i] * w1[j];
    }
}

// ---------- WMMA fp32 GEMM: out[N,64] = A[N,64] @ W[64,64] ----------
// One wave per 16x16 output tile; K=64 -> 16 chained V_WMMA_F32_16X16X4_F32.
__global__ void k_gemm_wmma(const float* __restrict__ A, const float* __restrict__ W,
                            float* __restrict__ Out, int nrows) {
    int gtid  = blockIdx.x * blockDim.x + threadIdx.x;
    int wave  = gtid >> 5;
    int lane  = threadIdx.x & 31;
    int tileM = wave >> 2;          // 4 column tiles (H=64)
    int tileN = wave & 3;
    if (tileM * 16 >= nrows) return;            // wave-uniform: EXEC stays all-1s

    const int m0    = tileM * 16;
    const int n0    = tileN * 16;
    const int mlane = lane & 15;
    const int koff  = (lane < 16) ? 0 : 2;      // A/B lane-half K offset per ISA layout

    const float* __restrict__ arow = A + (size_t)(m0 + mlane) * HID;

    v8f c = {};
#pragma unroll
    for (int k = 0; k < 16; ++k) {
        v2f a = *(const v2f*)(arow + 4 * k + koff);           // K = 4k+koff, 4k+koff+1
        v2f b;
        b.x = W[(4 * k + koff)     * HID + n0 + mlane];
        b.y = W[(4 * k + koff + 1) * HID + n0 + mlane];
        c = __builtin_amdgcn_wmma_f32_16x16x4_f32(
                /*neg_a=*/false, a, /*neg_b=*/false, b,
                /*c_mod=*/(short)0, c, /*reuse_a=*/false, /*reuse_b=*/false);
    }

    const int rbase = m0 + ((lane < 16) ? 0 : 8);             // C/D layout: VGPR v -> row v / v+8
#pragma unroll
    for (int v = 0; v < 8; ++v) {
        Out[(size_t)(rbase + v) * HID + n0 + mlane] = c[v];
    }
}

// ---------- self-loop init: out = dinv[i]^2 * t (also initializes scatter target) ----------
__global__ void k_selfloop(const float* __restrict__ dinv, const float* __restrict__ t,
                           float* __restrict__ out, int total) {
    int tid = blockIdx.x * blockDim.x + threadIdx.x;
    if (tid < total) {
        float di = dinv[tid >> 6];
        out[tid] = di * di * t[tid];
    }
}

// ---------- edge scatter: out[dst] += t[src] * norm  (16 lanes/edge, float4) ----------
__global__ void k_edge_scatter(const int* __restrict__ src, const int* __restrict__ dst,
                               const float* __restrict__ norm, const float* __restrict__ t,
                               float* __restrict__ out, int e) {
    int tid  = blockIdx.x * blockDim.x + threadIdx.x;
    int eidx = tid >> 4;
    if (eidx >= e) return;
    int   part = tid & 15;
    int   s = src[eidx], d = dst[eidx];
    float w = norm[eidx];
    const float4 v = reinterpret_cast<const float4*>(t + (size_t)s * HID)[part];
    float* o = out + (size_t)d * HID + part * 4;
    atomicAdd(o + 0, v.x * w);
    atomicAdd(o + 1, v.y * w);
    atomicAdd(o + 2, v.z * w);
    atomicAdd(o + 3, v.w * w);
}

// ---------- bias + relu (in place) ----------
__global__ void k_bias_relu(float* __restrict__ h, const float* __restrict__ b, int total) {
    int tid = blockIdx.x * blockDim.x + threadIdx.x;
    if (tid < total) {
        float v = h[tid] + b[tid & 63];
        h[tid] = (v > 0.0f) ? v : 0.0f;
    }
}

// ---------- global mean pool ----------
__global__ void k_zero_pool(float* __restrict__ pool, float* __restrict__ cnt) {
    int tid = blockIdx.x * blockDim.x + threadIdx.x;
    if (tid < NG * HID) pool[tid] = 0.0f;
    if (tid < NG)       cnt[tid]  = 0.0f;
}

__global__ void k_pool_accum(const int* __restrict__ batch, const float* __restrict__ h,
                             float* __restrict__ pool, float* __restrict__ cnt, int total) {
    int tid = blockIdx.x * blockDim.x + threadIdx.x;
    if (tid < total) {
        int i = tid >> 6, j = tid & 63;
        int g = batch[i];
        atomicAdd(&pool[(size_t)g * HID + j], h[tid]);
        if (j == 0) atomicAdd(&cnt[g], 1.0f);
    }
}

__global__ void k_final(const float* __restrict__ pool, const float* __restrict__ cnt,
                        const float* __restrict__ fc_w, const float* __restrict__ fc_b,
                        float* __restrict__ out) {
    int g = blockIdx.x * blockDim.x + threadIdx.x;
    if (g < NG) {
        float acc = 0.0f;
        const float* row = pool + (size_t)g * HID;
#pragma unroll 8
        for (int j = 0; j < HID; ++j) acc += row[j] * fc_w[j];
        float c = cnt[g];
        out[g] = acc / fmaxf(c, 1.0f) + fc_b[0];
    }
}

extern "C" void kernel_launch(void* const* d_in, const int* in_sizes, int n_in,
                              void* d_out, int out_size, void* d_ws, size_t ws_size,
                              hipStream_t stream) {
    const float* x     = (const float*)d_in[0];
    const int*   ei    = (const int*)  d_in[1];
    const int*   batch = (const int*)  d_in[2];
    const float* w1    = (const float*)d_in[3];
    const float* b1    = (const float*)d_in[4];
    const float* w2    = (const float*)d_in[5];
    const float* b2    = (const float*)d_in[6];
    const float* w3    = (const float*)d_in[7];
    const float* b3    = (const float*)d_in[8];
    const float* fcw   = (const float*)d_in[9];
    const float* fcb   = (const float*)d_in[10];
    float* out = (float*)d_out;

    const int* src = ei;        // edge_index[0]
    const int* dst = ei + NE;   // edge_index[1]

    // workspace carve-up (floats, 256B-aligned sections)
    float* dinv = (float*)d_ws;                                // N
    float* norm = dinv + ((NN + 63) & ~63);                    // E
    float* hA   = norm + ((NE + 63) & ~63);                    // N*64 (activations / agg out)
    float* hB   = hA + (size_t)NN * HID;                       // N*64 (transformed t)
    float* pool = hB + (size_t)NN * HID;                       // G*64
    float* cnt  = pool + (size_t)NG * HID;                     // G

    const int NH   = NN * HID;
    const int blkN = (NN + 255) / 256;
    const int blkE = (NE + 255) / 256;
    const int blkNH = (NH + 255) / 256;
    const int blkSc = (NE * 16 + 255) / 256;
    const int gemmWaves  = (NN / 16) * 4;                      // N divisible by 16
    const int gemmBlocks = (gemmWaves * 32 + 255) / 256;

    // normalization precompute
    k_init_deg  <<<blkN, 256, 0, stream>>>(dinv, NN);
    k_deg_accum <<<blkE, 256, 0, stream>>>(dst, dinv, NE);
    k_dinv      <<<blkN, 256, 0, stream>>>(dinv, NN);
    k_norm      <<<blkE, 256, 0, stream>>>(src, dst, dinv, norm, NE);

    // layer 1: t = x * w1 ; agg ; bias+relu
    k_outer1      <<<blkNH, 256, 0, stream>>>(x, w1, hB, NH);
    k_selfloop    <<<blkNH, 256, 0, stream>>>(dinv, hB, hA, NH);
    k_edge_scatter<<<blkSc, 256, 0, stream>>>(src, dst, norm, hB, hA, NE);
    k_bias_relu   <<<blkNH, 256, 0, stream>>>(hA, b1, NH);

    // layer 2: t = hA @ w2 (WMMA) ; agg ; bias+relu
    k_gemm_wmma   <<<gemmBlocks, 256, 0, stream>>>(hA, w2, hB, NN);
    k_selfloop    <<<blkNH, 256, 0, stream>>>(dinv, hB, hA, NH);
    k_edge_scatter<<<blkSc, 256, 0, stream>>>(src, dst, norm, hB, hA, NE);
    k_bias_relu   <<<blkNH, 256, 0, stream>>>(hA, b2, NH);

    // layer 3
    k_gemm_wmma   <<<gemmBlocks, 256, 0, stream>>>(hA, w3, hB, NN);
    k_selfloop    <<<blkNH, 256, 0, stream>>>(dinv, hB, hA, NH);
    k_edge_scatter<<<blkSc, 256, 0, stream>>>(src, dst, norm, hB, hA, NE);
    k_bias_relu   <<<blkNH, 256, 0, stream>>>(hA, b3, NH);

    // global mean pool + FC
    k_zero_pool <<<(NG * HID + 255) / 256, 256, 0, stream>>>(pool, cnt);
    k_pool_accum<<<blkNH, 256, 0, stream>>>(batch, hA, pool, cnt, NH);
    k_final     <<<(NG + 255) / 256, 256, 0, stream>>>(pool, cnt, fcw, fcb, out);
}